// GATBlock_47742856462353
// MI455X (gfx1250) — compile-verified
//
#include <hip/hip_runtime.h>
#include <math.h>

// ---------------------------------------------------------------------------
// Problem constants (from the reference)
// ---------------------------------------------------------------------------
#define NN      100000
#define EE      1600000
#define EP      (EE + NN)     // edges + self loops
#define DIM     128           // IN_DIM == OUT_DIM == HEADS*C
#define HEADS   4
#define CH      32            // channels per head
#define NEG     0.2f
#define BN_EPS  1e-5f

typedef __attribute__((ext_vector_type(2))) float v2f;
typedef __attribute__((ext_vector_type(8))) float v8f;

// ---------------------------------------------------------------------------
// Monotone float <-> uint key for atomicMax on floats
// ---------------------------------------------------------------------------
__device__ __forceinline__ unsigned fkey(float f) {
    unsigned u = __float_as_uint(f);
    return (u >> 31) ? ~u : (u | 0x80000000u);
}
__device__ __forceinline__ float funkey(unsigned k) {
    return __uint_as_float((k >> 31) ? (k & 0x7FFFFFFFu) : ~k);
}

// ---------------------------------------------------------------------------
// Kernel 1: h = x @ W  via V_WMMA_F32_16X16X4_F32 (fp32 WMMA, exact precision)
// One wave -> one 16x16 output tile. Block = 8 waves = the 8 column tiles of
// one 16-row strip. N is a multiple of 16, so no bounds divergence and EXEC
// stays all-ones across the WMMA (ISA requirement).
// ---------------------------------------------------------------------------
__global__ __launch_bounds__(256) void gemm_h_wmma(const float* __restrict__ x,
                                                   const float* __restrict__ W,
                                                   float* __restrict__ h) {
    const int wave = threadIdx.x >> 5;      // 0..7 : column tile
    const int lane = threadIdx.x & 31;
    const int half = lane >> 4;             // 0: K={0,1}, 1: K={2,3}
    const int l    = lane & 15;
    const int m0   = blockIdx.x << 4;       // row tile base
    const int n0   = wave << 4;             // col tile base

    const int rowA = m0 + l;                // A-matrix row for this lane
    const int colB = n0 + l;                // B-matrix col for this lane

    v8f acc = {};
#if defined(__HIP_DEVICE_COMPILE__)
    const float* __restrict__ xr = x + (size_t)rowA * DIM;
    for (int k = 0; k < DIM; k += 4) {
        const int ka = k + half * 2;
        v2f a; a.x = xr[ka];                 a.y = xr[ka + 1];            // A 16x4
        v2f b; b.x = W[ka * DIM + colB];     b.y = W[(ka + 1) * DIM + colB]; // B 4x16
        acc = __builtin_amdgcn_wmma_f32_16x16x4_f32(
                  false, a, false, b, (short)0, acc, false, false);
    }
#endif
    // C/D layout: lane<16 -> rows m0..m0+7 in v0..v7; lane>=16 -> rows m0+8..m0+15
    const int col   = n0 + l;
    const int rbase = m0 + half * 8;
#pragma unroll
    for (int r = 0; r < 8; ++r)
        h[(size_t)(rbase + r) * DIM + col] = acc[r];
}

// ---------------------------------------------------------------------------
// Kernel 2: per-(node,head) attention dot products
// ---------------------------------------------------------------------------
__global__ __launch_bounds__(256) void attn_logits(const float* __restrict__ h,
                                                   const float* __restrict__ att_src,
                                                   const float* __restrict__ att_dst,
                                                   float* __restrict__ a_src,
                                                   float* __restrict__ a_dst) {
    const int t = blockIdx.x * blockDim.x + threadIdx.x;
    if (t >= NN * HEADS) return;
    const int node = t >> 2, head = t & 3;
    const float* hp = h + (size_t)node * DIM + head * CH;
    const float* as = att_src + head * CH;
    const float* ad = att_dst + head * CH;
    float s = 0.f, d = 0.f;
#pragma unroll 8
    for (int c = 0; c < CH; ++c) { float v = hp[c]; s += v * as[c]; d += v * ad[c]; }
    a_src[t] = s;
    a_dst[t] = d;
}

// ---------------------------------------------------------------------------
// Kernel 3: segment max of leaky-relu logits (ordered-uint atomicMax)
// ---------------------------------------------------------------------------
__global__ __launch_bounds__(256) void edge_max(const int* __restrict__ ei,
                                                const float* __restrict__ a_src,
                                                const float* __restrict__ a_dst,
                                                unsigned* __restrict__ amax) {
    const int e = blockIdx.x * blockDim.x + threadIdx.x;
    if (e >= EP) return;
    int s, d;
    if (e < EE) { s = ei[e]; d = ei[EE + e]; } else { s = d = e - EE; }
#pragma unroll
    for (int hd = 0; hd < HEADS; ++hd) {
        float l = a_src[s * HEADS + hd] + a_dst[d * HEADS + hd];
        l = (l >= 0.f) ? l : NEG * l;
        atomicMax(&amax[d * HEADS + hd], fkey(l));
    }
}

// ---------------------------------------------------------------------------
// Kernel 4: segment sum of exp(l - max)
// ---------------------------------------------------------------------------
__global__ __launch_bounds__(256) void edge_sum(const int* __restrict__ ei,
                                                const float* __restrict__ a_src,
                                                const float* __restrict__ a_dst,
                                                const unsigned* __restrict__ amax,
                                                float* __restrict__ asum) {
    const int e = blockIdx.x * blockDim.x + threadIdx.x;
    if (e >= EP) return;
    int s, d;
    if (e < EE) { s = ei[e]; d = ei[EE + e]; } else { s = d = e - EE; }
#pragma unroll
    for (int hd = 0; hd < HEADS; ++hd) {
        float l = a_src[s * HEADS + hd] + a_dst[d * HEADS + hd];
        l = (l >= 0.f) ? l : NEG * l;
        float w = __expf(l - funkey(amax[d * HEADS + hd]));
        atomicAdd(&asum[d * HEADS + hd], w);
    }
}

// ---------------------------------------------------------------------------
// Kernel 5: weighted scatter-aggregate. One wave per edge: lane loads a
// coalesced float4 of h[src] (full 128-float row per wave), scales by its
// head's softmax coefficient, 4 coalesced f32 atomic adds into out[dst].
// out[] lives in d_out (51 MB -> L2 resident on MI455X).
// ---------------------------------------------------------------------------
__global__ __launch_bounds__(256) void edge_aggr(const int* __restrict__ ei,
                                                 const float* __restrict__ h,
                                                 const float* __restrict__ a_src,
                                                 const float* __restrict__ a_dst,
                                                 const unsigned* __restrict__ amax,
                                                 const float* __restrict__ asum,
                                                 float* __restrict__ out) {
    const long long t = (long long)blockIdx.x * blockDim.x + threadIdx.x;
    const long long e = t >> 5;
    const int lane = (int)(t & 31);
    if (e >= EP) return;

    // prefetch edge list ahead (lowers to global_prefetch_b8; speculative)
    if (e + 8192 < EE) __builtin_prefetch(ei + e + 8192, 0, 1);

    int s, d;
    if (e < EE) { s = ei[e]; d = ei[EE + e]; } else { s = d = (int)(e - EE); }

    const int hd = lane >> 3;               // 8 lanes per head (4 floats each)
    float l = a_src[s * HEADS + hd] + a_dst[d * HEADS + hd];
    l = (l >= 0.f) ? l : NEG * l;
    const float coef = __expf(l - funkey(amax[d * HEADS + hd])) / asum[d * HEADS + hd];

    const float4 hv = *(const float4*)(h + (size_t)s * DIM + lane * 4);
    float* op = out + (size_t)d * DIM + lane * 4;
    atomicAdd(op + 0, coef * hv.x);
    atomicAdd(op + 1, coef * hv.y);
    atomicAdd(op + 2, coef * hv.z);
    atomicAdd(op + 3, coef * hv.w);
}

// ---------------------------------------------------------------------------
// Kernel 6: BN column partial sums (thread = channel, coalesced rows)
// ---------------------------------------------------------------------------
__global__ __launch_bounds__(128) void bn_partial(const float* __restrict__ prev,
                                                  const float* __restrict__ msg,
                                                  const float* __restrict__ bias,
                                                  float* __restrict__ bnsum,
                                                  float* __restrict__ bnsumsq) {
    const int c = threadIdx.x;
    const float b = bias[c];
    float s = 0.f, ss = 0.f;
    for (int r = blockIdx.x; r < NN; r += gridDim.x) {
        float y = prev[(size_t)r * DIM + c] + msg[(size_t)r * DIM + c] + b;
        s += y; ss += y * y;
    }
    atomicAdd(&bnsum[c], s);
    atomicAdd(&bnsumsq[c], ss);
}

// ---------------------------------------------------------------------------
// Kernel 7: finalize BN stats -> per-channel scale/shift
// ---------------------------------------------------------------------------
__global__ void bn_finalize(const float* __restrict__ bnsum,
                            const float* __restrict__ bnsumsq,
                            const float* __restrict__ gamma,
                            const float* __restrict__ beta,
                            float* __restrict__ scale,
                            float* __restrict__ shift) {
    const int c = threadIdx.x;
    const float inv_n = 1.0f / (float)NN;
    const float mean = bnsum[c] * inv_n;
    const float var  = bnsumsq[c] * inv_n - mean * mean;
    const float sc   = gamma[c] * rsqrtf(var + BN_EPS);
    scale[c] = sc;
    shift[c] = beta[c] - mean * sc;
}

// ---------------------------------------------------------------------------
// Kernel 8: y = relu(scale*(prev + msg + bias) + shift), float4, in-place d_out
// ---------------------------------------------------------------------------
__global__ __launch_bounds__(256) void apply_bn(const float* __restrict__ prev,
                                                const float* __restrict__ bias,
                                                const float* __restrict__ scale,
                                                const float* __restrict__ shift,
                                                float* __restrict__ out) {
    const int t = blockIdx.x * blockDim.x + threadIdx.x;
    if (t >= NN * (DIM / 4)) return;
    const int r  = t >> 5;           // row
    const int c4 = t & 31;           // float4 column group
    const size_t base = (size_t)r * DIM + c4 * 4;
    const int c0 = c4 * 4;

    float4 p = *(const float4*)(prev + base);
    float4 m = *(const float4*)(out + base);   // message accumulator
    float v[4] = {p.x + m.x, p.y + m.y, p.z + m.z, p.w + m.w};
    float4 o;
    float* ov = (float*)&o;
#pragma unroll
    for (int i = 0; i < 4; ++i) {
        float y = (v[i] + bias[c0 + i]) * scale[c0 + i] + shift[c0 + i];
        ov[i] = y > 0.f ? y : 0.f;
    }
    *(float4*)(out + base) = o;
}

// ---------------------------------------------------------------------------
// Host launcher
// ---------------------------------------------------------------------------
extern "C" void kernel_launch(void* const* d_in, const int* in_sizes, int n_in,
                              void* d_out, int out_size, void* d_ws, size_t ws_size,
                              hipStream_t stream) {
    const float* prev    = (const float*)d_in[0];
    const float* x       = (const float*)d_in[1];
    const int*   ei      = (const int*)  d_in[2];
    const float* W       = (const float*)d_in[3];
    const float* att_src = (const float*)d_in[4];
    const float* att_dst = (const float*)d_in[5];
    const float* bias    = (const float*)d_in[6];
    const float* gamma   = (const float*)d_in[7];
    const float* beta    = (const float*)d_in[8];
    float* out = (float*)d_out;

    // workspace layout (floats)
    float*    h      = (float*)d_ws;                        // N*128
    float*    a_src  = h + (size_t)NN * DIM;                // N*4
    float*    a_dst  = a_src + (size_t)NN * HEADS;          // N*4
    unsigned* amax   = (unsigned*)(a_dst + (size_t)NN * HEADS); // N*4 (uint keys)
    float*    asum   = (float*)(amax + (size_t)NN * HEADS); // N*4
    float*    bnsum  = asum + (size_t)NN * HEADS;           // 128
    float*    bnsqsm = bnsum + DIM;                         // 128
    float*    scale  = bnsqsm + DIM;                        // 128
    float*    shift  = scale + DIM;                         // 128

    // zero the accumulators every call (deterministic across graph replays).
    // amax key 0 == "-inf"; asum/bnsum/bnsumsq need 0; d_out is the msg accum.
    hipMemsetAsync(amax, 0, ((size_t)NN * HEADS * 2 + 4 * DIM) * sizeof(float), stream);
    hipMemsetAsync(out, 0, (size_t)NN * DIM * sizeof(float), stream);

    // 1) GEMM h = x @ W  (fp32 WMMA): 6250 row strips x 8 column-tile waves
    gemm_h_wmma<<<NN / 16, 256, 0, stream>>>(x, W, h);

    // 2) attention logits
    attn_logits<<<(NN * HEADS + 255) / 256, 256, 0, stream>>>(h, att_src, att_dst, a_src, a_dst);

    // 3-4) segment softmax stats over edges (+ self loops)
    edge_max<<<(EP + 255) / 256, 256, 0, stream>>>(ei, a_src, a_dst, amax);
    edge_sum<<<(EP + 255) / 256, 256, 0, stream>>>(ei, a_src, a_dst, amax, asum);

    // 5) weighted scatter-aggregate: one wave per edge
    {
        long long threads = (long long)EP * 32;
        int blocks = (int)((threads + 255) / 256);
        edge_aggr<<<blocks, 256, 0, stream>>>(ei, h, a_src, a_dst, amax, asum, out);
    }

    // 6-8) residual + BatchNorm(training stats) + ReLU
    bn_partial<<<240, 128, 0, stream>>>(prev, out, bias, bnsum, bnsqsm);
    bn_finalize<<<1, 128, 0, stream>>>(bnsum, bnsqsm, gamma, beta, scale, shift);
    apply_bn<<<(NN * (DIM / 4) + 255) / 256, 256, 0, stream>>>(prev, bias, scale, shift, out);
}